// GATLayer_27719718928467
// MI455X (gfx1250) — compile-verified
//
#include <hip/hip_runtime.h>

typedef float v2f __attribute__((ext_vector_type(2)));
typedef float v8f __attribute__((ext_vector_type(8)));

#define Bc   8
#define Lc   2048
#define Ec   4096
#define TD   768
#define ED   256
#define HID  128
#define LDA  772   // padded LDS row stride (dwords): 772 % 64 == 4 -> conflict-free
#define KQ   (TD / 4)   // 192 K-quads

// ---------------- ws layout (float offsets) ----------------
#define OFF_PE    0            // edge partial sums: 8*16*256
#define OFF_BIAS  32768        // bias: 8*128
#define OFF_H     33792        // h: 16384*128
#define OFF_W     2130944      // exp weights: 8*2048
#define OFF_INVZ  2147328      // 1/Z per batch: 8
#define OFF_PW    2147336      // weighted partials: 8*16*128
#define OFF_OV    2163720      // out vectors: 8*128
#define OFF_WSW   2164744      // fragment-major W_fc[:768]: 8*192*64 = 98304

// 1) Partial column sums of edge_embedding over 256-row chunks.
__global__ void __launch_bounds__(256) k_edge_partial(const float* __restrict__ edge,
                                                      float* __restrict__ pe) {
    int b = blockIdx.x, rc = blockIdx.y, c = threadIdx.x;
    const float* p = edge + ((size_t)b * Ec + (size_t)rc * 256) * ED + c;
    float s = 0.f;
#pragma unroll 8
    for (int r = 0; r < 256; ++r) s += p[(size_t)r * ED];
    pe[(b * 16 + rc) * ED + c] = s;
}

// 2) avg_edge (in LDS) -> per-batch bias = avg_edge @ W_fc[768:,:]
__global__ void __launch_bounds__(256) k_avg_bias(const float* __restrict__ pe,
                                                  const float* __restrict__ Wfc,
                                                  float* __restrict__ bias) {
    __shared__ float avg[ED];
    int b = blockIdx.x, t = threadIdx.x;
    float s = 0.f;
#pragma unroll
    for (int rc = 0; rc < 16; ++rc) s += pe[(b * 16 + rc) * ED + t];
    avg[t] = s * (1.0f / (float)Ec);
    __syncthreads();
    if (t < HID) {
        float acc = 0.f;
#pragma unroll 4
        for (int c = 0; c < ED; ++c)
            acc += avg[c] * Wfc[(size_t)(TD + c) * HID + t];
        bias[b * HID + t] = acc;
    }
}

// 2b) Repack W_fc[:768,:] into fragment-major order:
//     wsw[((ntile*KQ + kq)*32 + lane)*2 + j] = Wfc[(kq*4 + (lane>>4)*2 + j)*128 + ntile*16 + (lane&15)]
//     -> each WMMA B fragment becomes one coalesced global_load_b64 (256B/record).
__global__ void __launch_bounds__(256) k_repack(const float* __restrict__ Wfc,
                                                float* __restrict__ wsw) {
    int idx = blockIdx.x * 256 + threadIdx.x;        // 0 .. 98303
    int j     = idx & 1;
    int lane  = (idx >> 1) & 31;
    int kq    = (idx >> 6) % KQ;
    int ntile = (idx >> 6) / KQ;
    int k = kq * 4 + ((lane >> 4) * 2) + j;
    int n = ntile * 16 + (lane & 15);
    wsw[idx] = Wfc[(size_t)k * HID + n];
}

// 3) h = token @ W_fc[:768,:] + bias[b]  via V_WMMA_F32_16X16X4_F32.
//    One block = one 16-row M tile; 8 waves each own one 16-col N tile.
__global__ void __launch_bounds__(256) k_gemm_wmma(const float* __restrict__ tok,
                                                   const float* __restrict__ wsw,
                                                   const float* __restrict__ bias,
                                                   float* __restrict__ h) {
    __shared__ float As[16 * LDA];
    const int m0  = blockIdx.x * 16;
    const int b   = m0 >> 11;          // m0 / 2048
    const int tid = threadIdx.x;

    // Stage 16x768 A tile into LDS (float4, fully coalesced).
    const float4* tg = (const float4*)(tok + (size_t)m0 * TD);
    for (int i = tid; i < 16 * TD / 4; i += 256) {
        int r = (i * 4) / TD, c = (i * 4) % TD;
        *(float4*)(&As[r * LDA + c]) = tg[i];
    }
    __syncthreads();

    const int wave = tid >> 5, lane = tid & 31;
    const int mr = lane & 15;          // A: row M ; B/D: col N
    const int kh = (lane >> 4) * 2;    // K sub-pair select per lane half

    v8f acc = {0.f, 0.f, 0.f, 0.f, 0.f, 0.f, 0.f, 0.f};

    const float* Ap = &As[mr * LDA + kh];
    const float* Bp = wsw + (size_t)wave * KQ * 64 + lane * 2;

#pragma unroll 8
    for (int kq = 0; kq < KQ; ++kq) {
        v2f a  = *(const v2f*)(Ap + kq * 4);     // ds_load_b64 (merged to 2addr)
        v2f bb = *(const v2f*)(Bp + kq * 64);    // coalesced global_load_b64
        acc = __builtin_amdgcn_wmma_f32_16x16x4_f32(
            /*neg_a=*/false, a, /*neg_b=*/false, bb,
            /*c_mod=*/(short)0, acc, /*reuse_a=*/false, /*reuse_b=*/false);
    }

    const int col   = wave * 16 + mr;
    const float bv  = bias[b * HID + col];
    const int rbase = m0 + (lane >> 4) * 8;
#pragma unroll
    for (int r = 0; r < 8; ++r)
        h[(size_t)(rbase + r) * HID + col] = acc[r] + bv;
}

// 4) Per batch: s_j = h . a1, softmax weights exp(s_j - max) and 1/Z.
//    (s_i and b_attn cancel inside the row softmax.)
__global__ void __launch_bounds__(256) k_softmax(const float* __restrict__ h,
                                                 const float* __restrict__ a_attn,
                                                 float* __restrict__ wexp,
                                                 float* __restrict__ invZ) {
    __shared__ float sL[Lc];
    __shared__ float red[256];
    const int b = blockIdx.x, tid = threadIdx.x;
    const int wave = tid >> 5, lane = tid & 31;

    float4 a1 = *(const float4*)(a_attn + lane * 4);
    const float* hb = h + (size_t)b * Lc * HID;

    for (int j = wave; j < Lc; j += 8) {
        float4 hv = *(const float4*)(hb + (size_t)j * HID + lane * 4);
        float d = hv.x * a1.x + hv.y * a1.y + hv.z * a1.z + hv.w * a1.w;
#pragma unroll
        for (int off = 16; off > 0; off >>= 1) d += __shfl_xor(d, off, 32);
        if (lane == 0) sL[j] = d;
    }
    __syncthreads();

    float m = -3.402823466e38f;
    for (int j = tid; j < Lc; j += 256) m = fmaxf(m, sL[j]);
    red[tid] = m; __syncthreads();
    for (int s2 = 128; s2 > 0; s2 >>= 1) {
        if (tid < s2) red[tid] = fmaxf(red[tid], red[tid + s2]);
        __syncthreads();
    }
    m = red[0]; __syncthreads();

    float sum = 0.f;
    for (int j = tid; j < Lc; j += 256) {
        float e = expf(sL[j] - m);
        wexp[b * Lc + j] = e;
        sum += e;
    }
    red[tid] = sum; __syncthreads();
    for (int s2 = 128; s2 > 0; s2 >>= 1) {
        if (tid < s2) red[tid] += red[tid + s2];
        __syncthreads();
    }
    if (tid == 0) invZ[b] = 1.0f / red[0];
}

// 5) Weighted partial sums over 128-row chunks: pw = (1/Z) * sum_j w_j * h[j,:]
__global__ void __launch_bounds__(128) k_wsum(const float* __restrict__ h,
                                              const float* __restrict__ wexp,
                                              const float* __restrict__ invZ,
                                              float* __restrict__ pw) {
    const int b = blockIdx.x, jc = blockIdx.y, n = threadIdx.x;
    const float* hb = h + ((size_t)b * Lc + (size_t)jc * 128) * HID + n;
    const float* wb = wexp + b * Lc + jc * 128;  // uniform across lanes -> scalar loads
    float acc = 0.f;
#pragma unroll 4
    for (int j = 0; j < 128; ++j)
        acc += wb[j] * hb[(size_t)j * HID];
    pw[(b * 16 + jc) * HID + n] = acc * invZ[b];
}

// 6) Combine 16 partials -> out_vec[b][128]
__global__ void __launch_bounds__(128) k_combine(const float* __restrict__ pw,
                                                 float* __restrict__ ov) {
    const int b = blockIdx.x, n = threadIdx.x;
    float s = 0.f;
#pragma unroll
    for (int jc = 0; jc < 16; ++jc) s += pw[(b * 16 + jc) * HID + n];
    ov[b * HID + n] = s;
}

// 7) Broadcast out_vec to all L rows (8 MB, float4 stores).
__global__ void __launch_bounds__(256) k_bcast(const float* __restrict__ ov,
                                               float4* __restrict__ out) {
    size_t f = (size_t)blockIdx.x * 256 + threadIdx.x;  // float4 index
    int b  = (int)(f >> 16);   // f / (2048*32)
    int c4 = (int)(f & 31);
    out[f] = *(const float4*)(ov + b * HID + c4 * 4);
}

extern "C" void kernel_launch(void* const* d_in, const int* in_sizes, int n_in,
                              void* d_out, int out_size, void* d_ws, size_t ws_size,
                              hipStream_t stream) {
    const float* tok   = (const float*)d_in[0];  // (8,2048,768)
    const float* edge  = (const float*)d_in[1];  // (8,4096,256)
    const float* Wfc   = (const float*)d_in[2];  // (1024,128)
    const float* aatt  = (const float*)d_in[3];  // (256) ; only a1 = first 128 used
    // d_in[4] (b_attn) cancels in the row softmax -> unused.

    float* ws  = (float*)d_ws;
    float* out = (float*)d_out;

    k_edge_partial<<<dim3(Bc, 16), 256, 0, stream>>>(edge, ws + OFF_PE);
    k_avg_bias    <<<Bc, 256, 0, stream>>>(ws + OFF_PE, Wfc, ws + OFF_BIAS);
    k_repack      <<<(8 * KQ * 64) / 256, 256, 0, stream>>>(Wfc, ws + OFF_WSW);
    k_gemm_wmma   <<<(Bc * Lc) / 16, 256, 0, stream>>>(tok, ws + OFF_WSW, ws + OFF_BIAS, ws + OFF_H);
    k_softmax     <<<Bc, 256, 0, stream>>>(ws + OFF_H, aatt, ws + OFF_W, ws + OFF_INVZ);
    k_wsum        <<<dim3(Bc, 16), 128, 0, stream>>>(ws + OFF_H, ws + OFF_W, ws + OFF_INVZ, ws + OFF_PW);
    k_combine     <<<Bc, 128, 0, stream>>>(ws + OFF_PW, ws + OFF_OV);
    k_bcast       <<<(Bc * Lc * HID) / 4 / 256, 256, 0, stream>>>(ws + OFF_OV, (float4*)out);
}